// DiscreteContinuousEncoder_9191230014187
// MI455X (gfx1250) — compile-verified
//
#include <hip/hip_runtime.h>

// DISCO encoder: x[1,16,721,1440] (f32) -> y[1,16,361,720] (f32)
// Two chained f32 WMMA GEMMs per (ho, 16-wo) tile:
//   stage A: xk[k, wo]   = psi[k,(l,d)] @ xg[(l,d), wo]      (16x64x16, per channel)
//   stage B: y[o, wo]   += W[o,(c,k)]   @ xk[k, wo]          (16x12x16, per channel)
// All hot-loop LDS addresses are base-register + compile-time immediate.

#define CIN   16
#define COUT  16
#define KK    9
#define NL    7
#define ND    9
#define HIN   721
#define WIN   1440
#define HOUT  361
#define WOUT  720
#define PSC   2

#define WAVES   3
#define TW      16                          // wo columns per wave
#define TILE_W  (WAVES * TW)                // 48 wo columns per block (720 = 15*48)
#define WINW    (PSC * TILE_W + (ND - 1))   // 104-lon input window per block
#define LD      (NL * ND)                   // 63 real reduction terms (padded to 64)
#define CSTR    (NL * WINW)                 // 728: per-channel slab in s_xg

typedef __attribute__((ext_vector_type(2))) float v2f;
typedef __attribute__((ext_vector_type(8))) float v8f;

__global__ __launch_bounds__(WAVES * 32)
void disco_enc_kernel(const float* __restrict__ x,
                      const float* __restrict__ psi,
                      const float* __restrict__ wgt,
                      const float* __restrict__ quad,
                      const int*   __restrict__ lat_idx,
                      float* __restrict__ y)
{
    __shared__ float s_xg[CIN * CSTR];          // quad-weighted, lat-gathered window
    __shared__ float s_w[COUT * CIN * KK + 16]; // mixing weights (+ zero pad)
    __shared__ float s_xk[WAVES * 256];         // per-wave D->B fragment staging

    const int tid  = threadIdx.x;
    const int w    = tid >> 5;       // wave id in block
    const int lane = tid & 31;
    const int half = lane >> 4;      // lane half: selects K-pair / M-offset
    const int n    = lane & 15;      // N column (wo) / M row (A matrices)

    const int ho     = blockIdx.y;
    const int wo_blk = blockIdx.x * TILE_W;
    const int lon0   = (PSC * wo_blk - (ND / 2) + WIN) % WIN;   // window origin

    // ---- stage 0: gather input rows, apply quadrature, stage into LDS ----
#pragma unroll
    for (int l = 0; l < NL; ++l) {
        const int   row = lat_idx[ho * NL + l];   // uniform -> scalar load
        const float qw  = quad[row];
        for (int i = tid; i < CIN * WINW; i += WAVES * 32) {
            int c    = i / WINW;
            int lonl = i - c * WINW;
            int glon = lon0 + lonl; if (glon >= WIN) glon -= WIN;
            s_xg[c * CSTR + l * WINW + lonl] = x[(c * HIN + row) * WIN + glon] * qw;
        }
    }
    for (int i = tid; i < COUT * CIN * KK + 16; i += WAVES * 32)
        s_w[i] = (i < COUT * CIN * KK) ? wgt[i] : 0.0f;

    // ---- psi A-fragments: A[m=k, K=(l,d)], 16 K-steps, kept in 32 VGPRs ----
    // f32 16x4 A layout: lane = M row; VGPR0 -> K{0,2}, VGPR1 -> K{1,3} by half
    float a0[16], a1[16];
    {
        const int m = n;  // basis index (rows 9..15 zero-padded)
#pragma unroll
        for (int s = 0; s < 16; ++s) {
            int K0 = 4 * s + 2 * half;
            int K1 = K0 + 1;
            a0[s] = (m < KK && K0 < LD)
                  ? psi[((m * HOUT + ho) * NL + (K0 / ND)) * ND + (K0 % ND)] : 0.0f;
            a1[s] = (m < KK && K1 < LD)
                  ? psi[((m * HOUT + ho) * NL + (K1 / ND)) * ND + (K1 % ND)] : 0.0f;
        }
    }
    __syncthreads();

    // ---- loop-invariant base registers (element indices) ----
    const int wn    = w * TW + n;            // column inside block window
    const int baseh = PSC * wn + 2 * half;   // stage-A B: same-row K+2 for upper half
    const int base0 = PSC * wn;              // stage-A B: K=63 pad slot (A is 0 there)
    const int wbase = n * (CIN * KK) + 2 * half;      // stage-B A base into s_w
    const int xkb   = w * 256 + n + 32 * half;        // stage-B B base into s_xk
    const int xks   = w * 256 + n + 128 * half;       // stage-A D store base

    // 7 stage-A elements cross an l-row boundary between lane halves:
    // precompute their per-lane selected base (cndmask once, in the prologue).
    int crossSel[16][2];
#pragma unroll
    for (int s = 0; s < 16; ++s)
#pragma unroll
        for (int j = 0; j < 2; ++j) {
            const int KLo = 4 * s + j, KHi = KLo + 2;
            const int dLo = KLo % ND, lLo = KLo / ND;
            if (KHi < LD && dLo >= ND - 2) {
                const int dHi = KHi % ND, lHi = KHi / ND;
                crossSel[s][j] = PSC * wn + (half ? (lHi * WINW + dHi)
                                                  : (lLo * WINW + dLo));
            } else {
                crossSel[s][j] = 0;   // unused instance, DCE'd
            }
        }

    v8f acc = {};  // y D-fragment: rows o (= r + 8*half), cols wo (= n)

#pragma unroll
    for (int c = 0; c < CIN; ++c) {
        // ---- stage A: 16 x K=4 WMMA steps over the (l,d) support ----
        v8f xk = {};
#pragma unroll
        for (int s = 0; s < 16; ++s) {
            v2f a; a[0] = a0[s]; a[1] = a1[s];
            v2f b;
#pragma unroll
            for (int j = 0; j < 2; ++j) {
                const int KLo = 4 * s + j, KHi = KLo + 2;
                const int dLo = KLo % ND, lLo = KLo / ND;
                float v;
                if (KHi >= LD)            // K=63 pad: upper A element is 0
                    v = s_xg[base0 + c * CSTR + lLo * WINW + dLo];
                else if (dLo >= ND - 2)   // crosses l row between halves
                    v = s_xg[crossSel[s][j] + c * CSTR];
                else                      // common case: base + immediate
                    v = s_xg[baseh + c * CSTR + lLo * WINW + dLo];
                b[j] = v;
            }
            xk = __builtin_amdgcn_wmma_f32_16x16x4_f32(
                     false, a, false, b, (short)0, xk, false, false);
        }

        // ---- D -> B staging through per-wave LDS (same-wave DS is in-order) ----
#pragma unroll
        for (int r = 0; r < 8; ++r)
            s_xk[xks + 16 * r] = xk[r];

        // ---- stage B: 3 x K=4 WMMA steps over the 9 basis functions ----
        // (s_xk rows 9..15 are exact zeros, so k>=9 weight reads need no mask)
#pragma unroll
        for (int s0 = 0; s0 < 3; ++s0) {
            v2f a, b;
            a[0] = s_w[wbase + c * KK + 4 * s0 + 0];
            a[1] = s_w[wbase + c * KK + 4 * s0 + 1];
            b[0] = s_xk[xkb + 16 * (4 * s0 + 0)];
            b[1] = s_xk[xkb + 16 * (4 * s0 + 1)];
            acc = __builtin_amdgcn_wmma_f32_16x16x4_f32(
                      false, a, false, b, (short)0, acc, false, false);
        }
    }

    // ---- write y tile: D layout rows o = r + 8*half, col wo = n ----
#pragma unroll
    for (int r = 0; r < 8; ++r) {
        int o  = r + 8 * half;
        int wo = wo_blk + w * TW + n;
        y[(o * HOUT + ho) * WOUT + wo] = acc[r];
    }
}

extern "C" void kernel_launch(void* const* d_in, const int* in_sizes, int n_in,
                              void* d_out, int out_size, void* d_ws, size_t ws_size,
                              hipStream_t stream)
{
    (void)in_sizes; (void)n_in; (void)d_ws; (void)ws_size; (void)out_size;
    const float* x       = (const float*)d_in[0];   // [1,16,721,1440]
    const float* psi     = (const float*)d_in[1];   // [9,361,7,9]
    const float* wgt     = (const float*)d_in[2];   // [16,16,9]
    const float* quad    = (const float*)d_in[3];   // [721]
    const int*   lat_idx = (const int*)d_in[4];     // [361,7]
    float*       y       = (float*)d_out;           // [1,16,361,720]

    dim3 grid(WOUT / TILE_W, HOUT, 1);   // 15 x 361
    dim3 block(WAVES * 32, 1, 1);        // 96 threads = 3 waves
    disco_enc_kernel<<<grid, block, 0, stream>>>(x, psi, wgt, quad, lat_idx, y);
}